// QuantAct_10625749090695
// MI455X (gfx1250) — compile-verified
//
#include <hip/hip_runtime.h>
#include <cstdint>

// Problem constants (from reference setup_inputs): B=8, S=4096, D=768
#define NTOT   25165824    // 8*4096*768
#define DCH    768
#define NV4    (NTOT / 4)  // 6291456 float4 elements
#define F32EPS 1.1920928955078125e-07f

// ---- gfx1250 async-to-LDS path (guarded; falls back to plain LDS fill) ----
#if defined(__has_builtin)
#if __has_builtin(__builtin_amdgcn_global_load_async_to_lds_b32) && \
    __has_builtin(__builtin_amdgcn_s_wait_asynccnt)
#define USE_ASYNC_LDS 1
#endif
#endif
#ifndef USE_ASYNC_LDS
#define USE_ASYNC_LDS 0
#endif

#if USE_ASYNC_LDS
typedef __attribute__((address_space(1))) int g_i32;  // global (AS1) int*
typedef __attribute__((address_space(3))) int l_i32;  // LDS    (AS3) int*
static __device__ __forceinline__ void cp_async_dw(const void* gsrc,
                                                   void*       ldst) {
  __builtin_amdgcn_global_load_async_to_lds_b32(
      (g_i32*)(uintptr_t)gsrc,
      (l_i32*)(unsigned int)(uintptr_t)ldst,
      /*offset=*/0, /*cpol=*/0);
}
#endif

// ---------------- K0: init min/max/zscale cells in workspace ----------------
__global__ void k_init(float* __restrict__ wsf) {
  wsf[0] = INFINITY;   // running min of x+identity
  wsf[1] = -INFINITY;  // running max
  wsf[2] = 0.0f;       // z_scale (filled by k_scale)
}

// ---------------- K1: global min/max of (x + identity) ----------------------
__global__ __launch_bounds__(256) void k_minmax(const float4* __restrict__ X,
                                                const float4* __restrict__ I,
                                                float* __restrict__ wsf) {
  __shared__ float smin[256];
  __shared__ float smax[256];
  const unsigned tid = threadIdx.x;
  float lmin = INFINITY, lmax = -INFINITY;
  for (unsigned v = blockIdx.x * blockDim.x + tid; v < NV4;
       v += gridDim.x * blockDim.x) {
    float4 a = X[v];
    float4 b = I[v];
    float s0 = a.x + b.x, s1 = a.y + b.y, s2 = a.z + b.z, s3 = a.w + b.w;
    lmin = fminf(lmin, fminf(fminf(s0, s1), fminf(s2, s3)));
    lmax = fmaxf(lmax, fmaxf(fmaxf(s0, s1), fmaxf(s2, s3)));
  }
  smin[tid] = lmin;
  smax[tid] = lmax;
  __syncthreads();
  for (unsigned s = 128; s > 0; s >>= 1) {
    if (tid < s) {
      smin[tid] = fminf(smin[tid], smin[tid + s]);
      smax[tid] = fmaxf(smax[tid], smax[tid + s]);
    }
    __syncthreads();
  }
  if (tid == 0) {
    atomicMin(&wsf[0], smin[0]);  // global_atomic_min_num_f32
    atomicMax(&wsf[1], smax[0]);  // global_atomic_max_num_f32
  }
}

// ---- K2: z_scale + exact per-channel frexp fixed-point factors (double) ----
__global__ __launch_bounds__(256) void k_scale(const float* __restrict__ pre,
                                               const float* __restrict__ ids,
                                               float* __restrict__ wsf,
                                               double* __restrict__ s0,
                                               double* __restrict__ s1,
                                               float* __restrict__ out_zs) {
  __shared__ float shzs;
  if (threadIdx.x == 0) {
    float minv = wsf[0];
    float maxv = wsf[1];
    float zs   = fmaxf(fmaxf(-minv, maxv) / 127.0f, F32EPS);
    wsf[2]     = zs;
    *out_zs    = zs;   // z_scale output (d_out[N])
    shzs       = zs;
  }
  __syncthreads();
  const double zs = (double)shzs;
  for (int d = threadIdx.x; d < DCH; d += blockDim.x) {
    {
      double r = (double)pre[d] / zs;
      int    e;
      double m  = frexp(r, &e);                    // r = m * 2^e, m in [0.5,1)
      double mi = floor(m * 2147483648.0 + 0.5);   // ROUND_HALF_UP of m*2^31
      s0[d]     = ldexp(mi, e - 31);               // m_int * 2^-(31-e), exact
    }
    {
      double r = (double)ids[d] / zs;
      int    e;
      double m  = frexp(r, &e);
      double mi = floor(m * 2147483648.0 + 0.5);
      s1[d]     = ldexp(mi, e - 31);
    }
  }
}

// ---------------- K3: main elementwise requantization pass ------------------
static __device__ __forceinline__ float qone(float xv, float iv, int d,
                                             const float* sPre,
                                             const float* sIds,
                                             const double* sS0,
                                             const double* sS1, float zs) {
  float  zi = rintf(xv / sPre[d]);                     // round(x / pre), f32 RNE
  float  wi = rintf(iv / sIds[d]);                     // round(identity / ids)
  double t  = rint((double)zi * sS0[d]) + rint((double)wi * sS1[d]);
  float  q  = fminf(fmaxf((float)t, -128.0f), 127.0f); // clip to [-n-1, n]
  return q * zs;
}

__global__ __launch_bounds__(256) void k_quant(const float4* __restrict__ X,
                                               const float4* __restrict__ I,
                                               const float* __restrict__ pre,
                                               const float* __restrict__ ids,
                                               const double* __restrict__ s0,
                                               const double* __restrict__ s1,
                                               const float* __restrict__ wsf,
                                               float4* __restrict__ O) {
  __shared__ float  sPre[DCH];
  __shared__ float  sIds[DCH];
  __shared__ double sS0[DCH];
  __shared__ double sS1[DCH];
  const unsigned tid = threadIdx.x;

#if USE_ASYNC_LDS
  for (int i = (int)tid; i < DCH; i += 256) {
    cp_async_dw((const unsigned int*)pre + i, (unsigned int*)&sPre[i]);
    cp_async_dw((const unsigned int*)ids + i, (unsigned int*)&sIds[i]);
  }
  for (int i = (int)tid; i < 2 * DCH; i += 256) {
    cp_async_dw((const unsigned int*)s0 + i, ((unsigned int*)sS0) + i);
    cp_async_dw((const unsigned int*)s1 + i, ((unsigned int*)sS1) + i);
  }
  __builtin_amdgcn_s_wait_asynccnt(0);
#else
  for (int i = (int)tid; i < DCH; i += 256) {
    sPre[i] = pre[i];
    sIds[i] = ids[i];
    sS0[i]  = s0[i];
    sS1[i]  = s1[i];
  }
#endif
  __syncthreads();

  const float zs = wsf[2];
  for (unsigned v = blockIdx.x * 256u + tid; v < NV4; v += gridDim.x * 256u) {
    float4 a = X[v];
    float4 b = I[v];
    // element index = 4*v; channel d0 = (4*v) % 768 = (v % 192) * 4, no wrap
    int    d0 = (int)(v % 192u) << 2;
    float4 o;
    o.x = qone(a.x, b.x, d0 + 0, sPre, sIds, sS0, sS1, zs);
    o.y = qone(a.y, b.y, d0 + 1, sPre, sIds, sS0, sS1, zs);
    o.z = qone(a.z, b.z, d0 + 2, sPre, sIds, sS0, sS1, zs);
    o.w = qone(a.w, b.w, d0 + 3, sPre, sIds, sS0, sS1, zs);
    O[v] = o;
  }
}

extern "C" void kernel_launch(void* const* d_in, const int* in_sizes, int n_in,
                              void* d_out, int out_size, void* d_ws,
                              size_t ws_size, hipStream_t stream) {
  (void)in_sizes; (void)n_in; (void)out_size; (void)ws_size;
  // setup_inputs order: x, pre_act_scaling_factor, identity, identity_scaling_factor
  const float* x   = (const float*)d_in[0];
  const float* pre = (const float*)d_in[1];
  const float* idn = (const float*)d_in[2];
  const float* ids = (const float*)d_in[3];
  float*       out = (float*)d_out;  // [N] quantized output, then [1] z_scale

  float*  wsf = (float*)d_ws;                                   // min,max,zs
  double* s0  = (double*)((char*)d_ws + 16);                    // 768 doubles
  double* s1  = (double*)((char*)d_ws + 16 + DCH * sizeof(double));

  k_init<<<1, 1, 0, stream>>>(wsf);
  k_minmax<<<2048, 256, 0, stream>>>((const float4*)x, (const float4*)idn, wsf);
  k_scale<<<1, 256, 0, stream>>>(pre, ids, wsf, s0, s1, out + NTOT);
  k_quant<<<4096, 256, 0, stream>>>((const float4*)x, (const float4*)idn, pre,
                                    ids, s0, s1, wsf, (float4*)out);
}